// Atoms_19739669692614
// MI455X (gfx1250) — compile-verified
//
#include <hip/hip_runtime.h>
#include <math.h>

#define PI_F 3.14159265358979323846f
#define NSAMP 32768
#define NPAIR 256
#define NBATCH 16
#define MIN_FQ (20.0f/11025.0f)
#define MAX_FQ (3000.0f/11025.0f)

typedef float v2f __attribute__((ext_vector_type(2)));
typedef float v8f __attribute__((ext_vector_type(8)));
typedef int   i32x2 __attribute__((ext_vector_type(2)));

#define DEV static __device__ __forceinline__

DEV float clip01f(float v){ return fminf(fmaxf(v, 0.0f), 1.0f); }

// Optional CDNA5 async load-to-LDS path (guarded; falls back to plain ld/st).
#if defined(__has_builtin)
#  if __has_builtin(__builtin_amdgcn_global_load_async_to_lds_b64)
#    define HAVE_ASYNC_LDS 1
#  endif
#  if __has_builtin(__builtin_amdgcn_s_wait_asynccnt)
#    define HAVE_WAIT_ASYNC 1
#  endif
#endif

DEV void wait_asynccnt0(){
#if defined(HAVE_WAIT_ASYNC)
  __builtin_amdgcn_s_wait_asynccnt(0);
#else
  asm volatile("s_wait_asynccnt 0x0" ::: "memory");
#endif
}

// ---------------------------------------------------------------------------
// Parameter prep: parse x, scan fine_env (128), analytic Gaussian-max params.
// ---------------------------------------------------------------------------
__global__ void k_prep(const float* __restrict__ x,
                       float* __restrict__ par,
                       float* __restrict__ env){
  int p = blockIdx.x;
  int t = threadIdx.x;                  // 128 threads
  const float* xp = x + (size_t)p*172;
  __shared__ float s[128];
  s[t] = clip01f(xp[36+t])*2.0f - 1.0f;
  __syncthreads();
  for(int off=1; off<128; off<<=1){     // Hillis-Steele inclusive scan
    float add = (t>=off) ? s[t-off] : 0.0f;
    __syncthreads();
    s[t] += add;
    __syncthreads();
  }
  env[(size_t)p*128 + t] = clip01f(s[t]);
  if (t==0){
    float* pp = par + (size_t)p*48;
    float means = clip01f(xp[0]);
    float stds  = clip01f(xp[1])*0.1f;
    float amp   = clip01f(xp[2]); amp *= amp;
    float f0    = clip01f(xp[3]); f0 *= f0;
    float f0r   = MIN_FQ + f0*(MAX_FQ - MIN_FQ);
    if (f0r > 1.0f) f0r = 0.0f;
    float loc   = fminf(fmaxf(means*(float)NSAMP, -(float)(NSAMP/2)), (float)NSAMP*1.5f);
    float scale = fminf(fmaxf((1e-8f + stds)*(float)NSAMP, 0.0f), (float)(NSAMP-1));
    float c0    = -logf(scale) - 0.918938533204672f;     // -log(scale)-0.5*log(2pi)
    float ts    = roundf(fminf(fmaxf(loc, 0.0f), (float)(NSAMP-1)));
    float zm    = (ts - loc)/scale;
    float den   = __expf(-0.5f*zm*zm + c0) + 1e-8f;      // analytic max_norm denom
    pp[0]=loc; pp[1]=scale; pp[2]=c0; pp[3]=1.0f/den; pp[4]=amp;
    #pragma unroll
    for(int h=0; h<8; ++h){
      pp[8+h]  = f0r * (clip01f(xp[4+h])*8.0f) * PI_F;   // freqs
      float af = clip01f(xp[164+h]);
      pp[16+h] = af*af;                                  // amp_factors
      pp[24+h] = logf(clip01f(xp[12+h]));                // log(mags): cumprod = exp(k*log m)
    }
    #pragma unroll
    for(int i=0;i<16;++i) pp[32+i] = clip01f(xp[20+i]);  // noise_coeff
  }
}

// ---------------------------------------------------------------------------
// Oscillator bank: res = sum_h sin(freq_h*(t+1)) * ampf_h * interp(mags^f)
// ---------------------------------------------------------------------------
__global__ void k_res(const float* __restrict__ par, float* __restrict__ res){
  int p = blockIdx.x;
  int t = blockIdx.y*blockDim.x + threadIdx.x;
  const float* pp = par + (size_t)p*48;
  float pos = ((float)t + 0.5f)*(1.0f/256.0f) - 0.5f;    // 128 frames -> 32768
  pos = fminf(fmaxf(pos, 0.0f), 127.0f);
  int i0 = (int)pos; int i1 = min(i0+1, 127);
  float w = pos - (float)i0;
  float acc = 0.0f;
  float tp1 = (float)(t+1);
  #pragma unroll
  for(int h=0; h<8; ++h){
    float lm = pp[24+h];
    float m  = __expf(lm*(float)(i0+1))*(1.0f-w) + __expf(lm*(float)(i1+1))*w;
    acc += __sinf(pp[8+h]*tp1) * pp[16+h] * m;
  }
  res[(size_t)p*NSAMP + t] = acc;
}

// ---------------------------------------------------------------------------
// Pack real noise pairs into complex (length-16384) for packed rfft.
// ---------------------------------------------------------------------------
__global__ void k_pack_noise(const float* __restrict__ noise, float2* __restrict__ W){
  int p = blockIdx.x;
  int m = blockIdx.y*blockDim.x + threadIdx.x;            // < 16384
  W[(size_t)p*16384 + m] = make_float2(noise[(size_t)p*NSAMP + 2*m],
                                       noise[(size_t)p*NSAMP + 2*m + 1]);
}

// ---------------------------------------------------------------------------
// Four-step FFT, stage 1: per-wave DFT-256 (stride N2) as radix-16x16 done
// with V_WMMA_F32_16X16X4_F32 complex matmuls; applies the outer W_n^{b*c}
// twiddle on store. n = 256*N2. sign=-1 forward, +1 inverse (unnormalized).
// ---------------------------------------------------------------------------
__global__ void __launch_bounds__(256)
k_fft256(const float2* __restrict__ src, float2* __restrict__ dst,
         int nsig, int N2, float sign){
  __shared__ float2 lsc[8*256];                 // 256-pt scratch per wave (16 KB)
  int wid  = threadIdx.x >> 5;
  int lane = threadIdx.x & 31;
  int task = blockIdx.x*8 + wid;                // one (signal, b) per wave
  if (task >= nsig*N2) return;                  // wave-uniform; EXEC stays full
  int s = task / N2;
  int b = task - s*N2;
  int n = N2 << 8;
  const float2* X = src + (size_t)s*n;
  float2*       Y = dst + (size_t)s*n;
  int lo = lane & 15, hi = lane >> 4;

  // A = 16x16 DFT matrix F[m][k] = W16^{m*k}; fragments per 16x16x4 A-layout.
  v2f Fr[4], Fi[4], nFi[4], Br[4], Bi[4];
  #pragma unroll
  for(int q=0;q<4;++q){
    #pragma unroll
    for(int v=0;v<2;++v){
      int a2 = q*4 + v + 2*hi;                  // K index
      float ang = sign * (PI_F/8.0f) * (float)((lo*a2) & 15);
      float sn, cs; __sincosf(ang, &sn, &cs);
      Fr[q][v] = cs; Fi[q][v] = sn; nFi[q][v] = -sn;
      float2 mv = X[(size_t)((lo + 16*a2)*N2 + b)];   // B[a2][a1]=x[N2*(a1+16*a2)+b]
      Br[q][v] = mv.x; Bi[q][v] = mv.y;
    }
  }

  v8f Sr = {0,0,0,0,0,0,0,0}, Si = {0,0,0,0,0,0,0,0};
  #pragma unroll
  for(int q=0;q<4;++q)
    Sr = __builtin_amdgcn_wmma_f32_16x16x4_f32(false, Fr[q],  false, Br[q], (short)0, Sr, false, false);
  #pragma unroll
  for(int q=0;q<4;++q)
    Sr = __builtin_amdgcn_wmma_f32_16x16x4_f32(false, nFi[q], false, Bi[q], (short)0, Sr, false, false);
  #pragma unroll
  for(int q=0;q<4;++q)
    Si = __builtin_amdgcn_wmma_f32_16x16x4_f32(false, Fr[q],  false, Bi[q], (short)0, Si, false, false);
  #pragma unroll
  for(int q=0;q<4;++q)
    Si = __builtin_amdgcn_wmma_f32_16x16x4_f32(false, Fi[q],  false, Br[q], (short)0, Si, false, false);

  // Mid-stage twiddle W256^{a1*c1}, then transpose through wave-local LDS.
  float2* L = lsc + wid*256;
  #pragma unroll
  for(int r=0;r<8;++r){
    int c1 = r + 8*hi, a1 = lo;
    float ang = sign * (PI_F/128.0f) * (float)((a1*c1) & 255);
    float sn, cs; __sincosf(ang, &sn, &cs);
    L[c1*16 + a1] = make_float2(Sr[r]*cs - Si[r]*sn, Sr[r]*sn + Si[r]*cs);
  }
  asm volatile("s_wait_dscnt 0" ::: "memory");  // wave-local LDS RAW fence

  v2f Cr[4], Ci[4];
  #pragma unroll
  for(int q=0;q<4;++q){
    #pragma unroll
    for(int v=0;v<2;++v){
      int a1 = q*4 + v + 2*hi;
      float2 t2 = L[lo*16 + a1];                // B2[a1][c1] = S1tw[c1][a1]
      Cr[q][v] = t2.x; Ci[q][v] = t2.y;
    }
  }
  v8f Tr = {0,0,0,0,0,0,0,0}, Ti = {0,0,0,0,0,0,0,0};
  #pragma unroll
  for(int q=0;q<4;++q)
    Tr = __builtin_amdgcn_wmma_f32_16x16x4_f32(false, Fr[q],  false, Cr[q], (short)0, Tr, false, false);
  #pragma unroll
  for(int q=0;q<4;++q)
    Tr = __builtin_amdgcn_wmma_f32_16x16x4_f32(false, nFi[q], false, Ci[q], (short)0, Tr, false, false);
  #pragma unroll
  for(int q=0;q<4;++q)
    Ti = __builtin_amdgcn_wmma_f32_16x16x4_f32(false, Fr[q],  false, Ci[q], (short)0, Ti, false, false);
  #pragma unroll
  for(int q=0;q<4;++q)
    Ti = __builtin_amdgcn_wmma_f32_16x16x4_f32(false, Fi[q],  false, Cr[q], (short)0, Ti, false, false);

  // Store T[c*N2+b] with four-step twiddle W_n^{b*c}.
  #pragma unroll
  for(int r=0;r<8;++r){
    int c  = lo + 16*r + 128*hi;                // c = c1 + 16*c2
    int bc = (b*c) % n;
    float ang = sign * 2.0f*PI_F * (float)bc / (float)n;
    float sn, cs; __sincosf(ang, &sn, &cs);
    Y[(size_t)c*N2 + b] = make_float2(Tr[r]*cs - Ti[r]*sn, Tr[r]*sn + Ti[r]*cs);
  }
}

// ---------------------------------------------------------------------------
// Four-step FFT, stage 2: length-N2 Stockham radix-2 in LDS (4 rows/block),
// output in natural order dst[256*d + c]. LDS fill uses CDNA5 async
// load-to-LDS (ASYNCcnt) when the builtin is available.
// ---------------------------------------------------------------------------
__global__ void k_fftN2(const float2* __restrict__ src, float2* __restrict__ dst,
                        int nsig, int N2, float sign){
  extern __shared__ float2 smv[];               // 2 * 4 * N2 complex
  int half  = N2 >> 1;
  int cslot = threadIdx.x / half;               // 4 rows per block
  int j     = threadIdx.x - cslot*half;
  int id    = blockIdx.x*4 + cslot;
  int s     = id >> 8;
  int c     = id & 255;
  int n     = N2 << 8;
  float2* bufA = smv;
  float2* bufB = smv + 4*N2;
  if (s < nsig){
    const float2* T = src + (size_t)s*n + (size_t)c*N2;
#if defined(HAVE_ASYNC_LDS)
    __builtin_amdgcn_global_load_async_to_lds_b64(
        (__attribute__((address_space(1))) i32x2*)(uintptr_t)(T + j),
        (__attribute__((address_space(3))) i32x2*)(uintptr_t)(bufA + cslot*N2 + j),
        0, 0);
    __builtin_amdgcn_global_load_async_to_lds_b64(
        (__attribute__((address_space(1))) i32x2*)(uintptr_t)(T + j + half),
        (__attribute__((address_space(3))) i32x2*)(uintptr_t)(bufA + cslot*N2 + j + half),
        0, 0);
    wait_asynccnt0();
#else
    bufA[cslot*N2 + j]        = T[j];
    bufA[cslot*N2 + j + half] = T[j + half];
#endif
  }
  __syncthreads();
  float2* cur = bufA; float2* nxt = bufB;
  for(int Ns=1; Ns<N2; Ns<<=1){
    float2 v0 = cur[cslot*N2 + j];
    float2 v1 = cur[cslot*N2 + j + half];
    int jm = j & (Ns-1);
    float ang = sign * PI_F * (float)jm / (float)Ns;
    float sn, cs; __sincosf(ang, &sn, &cs);
    float2 w1 = make_float2(v1.x*cs - v1.y*sn, v1.x*sn + v1.y*cs);
    int idxD = (j/Ns)*(Ns<<1) + jm;
    nxt[cslot*N2 + idxD]      = make_float2(v0.x + w1.x, v0.y + w1.y);
    nxt[cslot*N2 + idxD + Ns] = make_float2(v0.x - w1.x, v0.y - w1.y);
    __syncthreads();
    float2* tmp = cur; cur = nxt; nxt = tmp;
  }
  if (s < nsig){
    float2* O = dst + (size_t)s*n;
    O[(size_t)j*256 + c]        = cur[cslot*N2 + j];
    O[(size_t)(j+half)*256 + c] = cur[cslot*N2 + j + half];
  }
}

// ---------------------------------------------------------------------------
// Packed rfft unpack -> real spectral shape H (interp of 16 coeffs) -> repack.
// N = 16384 complex (real length 32768). Thread handles bins (k, N-k).
// ---------------------------------------------------------------------------
DEV float interpH(const float* __restrict__ nc, int k){
  float pos = ((float)k + 0.5f)*(16.0f/16385.0f) - 0.5f;
  pos = fminf(fmaxf(pos, 0.0f), 15.0f);
  int i0 = (int)pos; int i1 = min(i0+1, 15);
  float w = pos - (float)i0;
  return nc[i0]*(1.0f-w) + nc[i1]*w;
}

__global__ void k_shape(const float2* __restrict__ Z, float2* __restrict__ Z2,
                        const float* __restrict__ par){
  const int N = 16384;
  int p = blockIdx.x;
  int k = blockIdx.y*blockDim.x + threadIdx.x;
  if (k > N/2) return;
  const float* nc = par + (size_t)p*48 + 32;
  const float2* Zp = Z  + (size_t)p*N;
  float2*       Op = Z2 + (size_t)p*N;
  float2 zk  = Zp[k & (N-1)];
  float2 znk = Zp[(N-k) & (N-1)];
  float2 Ze = make_float2(0.5f*(zk.x+znk.x), 0.5f*(zk.y-znk.y));
  float2 Zo = make_float2(0.5f*(zk.y+znk.y), 0.5f*(znk.x-zk.x));
  float ang = -PI_F*(float)k/(float)N;
  float sn, cs; __sincosf(ang, &sn, &cs);          // Wk = e^{-i*pi*k/N}
  float2 WZo = make_float2(cs*Zo.x - sn*Zo.y, cs*Zo.y + sn*Zo.x);
  float2 Xk  = make_float2(Ze.x + WZo.x,  Ze.y + WZo.y);
  float2 Xnk = make_float2(Ze.x - WZo.x, -(Ze.y - WZo.y));
  float Hk  = interpH(nc, k);
  float Hnk = interpH(nc, N-k);
  float2 Yk  = make_float2(Xk.x*Hk,  Xk.y*Hk);
  float2 Ynk = make_float2(Xnk.x*Hnk, Xnk.y*Hnk);
  float2 Ye = make_float2(0.5f*(Yk.x+Ynk.x), 0.5f*(Yk.y-Ynk.y));
  float2 D  = make_float2(0.5f*(Yk.x-Ynk.x), 0.5f*(Yk.y+Ynk.y));
  float2 Yo = make_float2(cs*D.x + sn*D.y, cs*D.y - sn*D.x);   // conj(Wk)*D
  Op[k & (N-1)]     = make_float2(Ye.x - Yo.y, Ye.y + Yo.x);   // Ye + i*Yo
  Op[(N-k) & (N-1)] = make_float2(Ye.x + Yo.y, Yo.x - Ye.y);   // conj(Ye)+i*conj(Yo)
}

__global__ void k_unpack_shaped(const float2* __restrict__ Z, float* __restrict__ sa){
  int p = blockIdx.x;
  int m = blockIdx.y*blockDim.x + threadIdx.x;
  // ortho^2 (1/M * M from backward inverse) cancels; only our unnormalized
  // sign=+1 complex inverse needs 1/N with N = 16384.
  const float sc = 1.0f/16384.0f;
  float2 v = Z[(size_t)p*16384 + m];
  sa[(size_t)p*NSAMP + 2*m]     = v.x*sc;
  sa[(size_t)p*NSAMP + 2*m + 1] = v.y*sc;
}

// atoms = probs * shaped * amp * fine_env  (in place on `sa`)
__global__ void k_atoms(const float* __restrict__ par, const float* __restrict__ env,
                        float* __restrict__ sa){
  int p = blockIdx.x;
  int t = blockIdx.y*blockDim.x + threadIdx.x;
  const float* pp = par + (size_t)p*48;
  float pos = ((float)t + 0.5f)*(1.0f/256.0f) - 0.5f;
  pos = fminf(fmaxf(pos, 0.0f), 127.0f);
  int i0 = (int)pos; int i1 = min(i0+1, 127);
  float w = pos - (float)i0;
  const float* ep = env + (size_t)p*128;
  float fe = ep[i0]*(1.0f-w) + ep[i1]*w;
  float z  = ((float)t - pp[0])/pp[1];
  float pv = __expf(-0.5f*z*z + pp[2]) * pp[3];
  size_t idx = (size_t)p*NSAMP + t;
  sa[idx] = pv * sa[idx] * pp[4] * fe;
}

// c[n] = atoms + i*res, optionally pre-twiddled by e^{-i*pi*n/32768} (odd bins)
__global__ void k_pack_conv(const float* __restrict__ sa, const float* __restrict__ res,
                            float2* __restrict__ W, int parity){
  int p = blockIdx.x;
  int t = blockIdx.y*blockDim.x + threadIdx.x;
  size_t idx = (size_t)p*NSAMP + t;
  float2 cc = make_float2(sa[idx], res[idx]);
  if (parity){
    float ang = -PI_F*(float)t*(1.0f/32768.0f);
    float sn, cs; __sincosf(ang, &sn, &cs);
    cc = make_float2(cc.x*cs - cc.y*sn, cc.x*sn + cc.y*cs);
  }
  W[idx] = cc;
}

// Per-bin A*B product (unpack packed atoms/res spectra), summed over 16 events.
__global__ void k_binprod(const float2* __restrict__ Ze, const float2* __restrict__ Zo,
                          float2* __restrict__ Sacc){
  int b = blockIdx.x;
  int m = blockIdx.y*blockDim.x + threadIdx.x;
  int mp = (NSAMP - m) & (NSAMP-1);
  int mo = (NSAMP-1) - m;
  float2 se = make_float2(0,0), so = make_float2(0,0);
  for(int e=0;e<16;++e){
    const float2* Z = Ze + (size_t)(b*16+e)*NSAMP;
    if (e < 15) __builtin_prefetch(Ze + (size_t)(b*16+e+1)*NSAMP + m, 0, 1);
    float2 c1 = Z[m], c2 = Z[mp];
    float2 A  = make_float2(0.5f*(c1.x+c2.x), 0.5f*(c1.y-c2.y));
    float2 Bv = make_float2(0.5f*(c1.y+c2.y), 0.5f*(c2.x-c1.x));
    se.x += A.x*Bv.x - A.y*Bv.y;
    se.y += A.x*Bv.y + A.y*Bv.x;
    const float2* Q = Zo + (size_t)(b*16+e)*NSAMP;
    c1 = Q[m]; c2 = Q[mo];
    A  = make_float2(0.5f*(c1.x+c2.x), 0.5f*(c1.y-c2.y));
    Bv = make_float2(0.5f*(c1.y+c2.y), 0.5f*(c2.x-c1.x));
    so.x += A.x*Bv.x - A.y*Bv.y;
    so.y += A.x*Bv.y + A.y*Bv.x;
  }
  Sacc[(size_t)b*NSAMP + m]      = se;   // even-bin products
  Sacc[(size_t)(16+b)*NSAMP + m] = so;   // odd-bin products
}

// y[n] = (1/65536) * Re( U[n] + e^{+i*pi*n/32768} * V[n] ), n < 32768
__global__ void k_final(const float2* __restrict__ Sacc, float* __restrict__ out){
  int b = blockIdx.x;
  int n = blockIdx.y*blockDim.x + threadIdx.x;
  float2 U = Sacc[(size_t)b*NSAMP + n];
  float2 V = Sacc[(size_t)(16+b)*NSAMP + n];
  float ang = PI_F*(float)n*(1.0f/32768.0f);
  float sn, cs; __sincosf(ang, &sn, &cs);
  out[(size_t)b*NSAMP + n] = (U.x + V.x*cs - V.y*sn) * (1.0f/65536.0f);
}

// ---------------------------------------------------------------------------
extern "C" void kernel_launch(void* const* d_in, const int* in_sizes, int n_in,
                              void* d_out, int out_size, void* d_ws, size_t ws_size,
                              hipStream_t stream) {
  const float* x     = (const float*)d_in[0];   // (16,16,172)
  const float* noise = (const float*)d_in[1];   // (16,16,32768)
  float* out = (float*)d_out;                   // (16,1,32768)
  float* ws  = (float*)d_ws;

  size_t oPar = 0;
  size_t oEnv = oPar + (size_t)NPAIR*48;
  size_t oRes = oEnv + (size_t)NPAIR*128;
  size_t oSA  = oRes + (size_t)NPAIR*NSAMP;
  size_t oW0  = oSA  + (size_t)NPAIR*NSAMP;
  size_t oW1  = oW0  + (size_t)NPAIR*NSAMP*2;
  size_t oW2  = oW1  + (size_t)NPAIR*NSAMP*2;
  size_t oAcc = oW2  + (size_t)NPAIR*NSAMP*2;
  float*  par  = ws + oPar;
  float*  env  = ws + oEnv;
  float*  res  = ws + oRes;
  float*  sa   = ws + oSA;
  float2* W0   = (float2*)(ws + oW0);
  float2* W1   = (float2*)(ws + oW1);
  float2* W2   = (float2*)(ws + oW2);
  float2* Sacc = (float2*)(ws + oAcc);

  k_prep<<<NPAIR, 128, 0, stream>>>(x, par, env);
  k_res <<<dim3(NPAIR,128), 256, 0, stream>>>(par, res);

  // --- noise shaping: packed rfft(32768) as complex FFT(16384) = 256 x 64 ---
  k_pack_noise<<<dim3(NPAIR,64), 256, 0, stream>>>(noise, W0);
  k_fft256<<<(NPAIR*64)/8, 256, 0, stream>>>(W0, W1, NPAIR, 64, -1.0f);
  k_fftN2 <<<(NPAIR*256)/4, 2*64, 2*4*64*sizeof(float2), stream>>>(W1, W2, NPAIR, 64, -1.0f);
  k_shape <<<dim3(NPAIR,33), 256, 0, stream>>>(W2, W0, par);
  k_fft256<<<(NPAIR*64)/8, 256, 0, stream>>>(W0, W1, NPAIR, 64,  1.0f);
  k_fftN2 <<<(NPAIR*256)/4, 2*64, 2*4*64*sizeof(float2), stream>>>(W1, W2, NPAIR, 64,  1.0f);
  k_unpack_shaped<<<dim3(NPAIR,64), 256, 0, stream>>>(W2, sa);
  k_atoms<<<dim3(NPAIR,128), 256, 0, stream>>>(par, env, sa);

  // --- conv: C65536(atoms + i*res) via even/odd pair of FFT(32768)=256x128 ---
  k_pack_conv<<<dim3(NPAIR,128), 256, 0, stream>>>(sa, res, W0, 0);
  k_fft256<<<(NPAIR*128)/8, 256, 0, stream>>>(W0, W1, NPAIR, 128, -1.0f);
  k_fftN2 <<<(NPAIR*256)/4, 2*128, 2*4*128*sizeof(float2), stream>>>(W1, W2, NPAIR, 128, -1.0f);  // Z_even
  k_pack_conv<<<dim3(NPAIR,128), 256, 0, stream>>>(sa, res, W0, 1);
  k_fft256<<<(NPAIR*128)/8, 256, 0, stream>>>(W0, W1, NPAIR, 128, -1.0f);
  k_fftN2 <<<(NPAIR*256)/4, 2*128, 2*4*128*sizeof(float2), stream>>>(W1, W0, NPAIR, 128, -1.0f);  // Z_odd

  // frequency-domain event sum (32 accumulator spectra: even|odd per batch)
  k_binprod<<<dim3(NBATCH,128), 256, 0, stream>>>(W2, W0, Sacc);

  // 32 inverse FFT(32768) instead of 512, then recombine even/odd halves
  k_fft256<<<(32*128)/8, 256, 0, stream>>>(Sacc, W1, 32, 128, 1.0f);
  k_fftN2 <<<(32*256)/4, 2*128, 2*4*128*sizeof(float2), stream>>>(W1, Sacc, 32, 128, 1.0f);
  k_final<<<dim3(NBATCH,128), 256, 0, stream>>>(Sacc, out);
}